// LinearizedMoeExperts_12283606466669
// MI455X (gfx1250) — compile-verified
//
#include <hip/hip_runtime.h>

// Problem constants (match reference)
#define T_TOK 4096
#define HDIM  2048
#define IDIM  1408
#define NEXP  8
#define TOPK  2
#define NPOS  (T_TOK * TOPK)        // 8192 routed rows
#define CAP   (NPOS + NEXP * 16)    // 8320: capacity with 16-row padding per expert

typedef __attribute__((ext_vector_type(2))) float v2f;
typedef __attribute__((ext_vector_type(8))) float v8f;

// Workspace layout (bytes)
#define OFF_COUNT   0                        // 8 ints
#define OFF_CURSOR  64                       // 8 ints
#define OFF_SEG     128                      // 9 ints (padded segment starts)
#define OFF_TOK     256                      // CAP ints
#define OFF_WEIGHT  (256 + CAP * 4)          // CAP floats
#define OFF_INTER   (OFF_WEIGHT + CAP * 4)   // CAP * IDIM floats (~46.9 MB)

// ---------------------------------------------------------------- routing ---

__global__ void zero_out_kernel(float4* __restrict__ out) {
  int i = blockIdx.x * blockDim.x + threadIdx.x;
  out[i] = make_float4(0.f, 0.f, 0.f, 0.f);   // grid exactly covers T*H/4
}

__global__ void route_init_kernel(int* __restrict__ cnt, int* __restrict__ cur,
                                  int* __restrict__ tok, float* __restrict__ wgt) {
  int i = blockIdx.x * blockDim.x + threadIdx.x;
  if (i < NEXP) { cnt[i] = 0; cur[i] = 0; }
  if (i < CAP)  { tok[i] = 0; wgt[i] = 0.f; }   // pad rows: token 0, weight 0
}

__global__ void route_count_kernel(const int* __restrict__ idx, int* __restrict__ cnt) {
  int i = blockIdx.x * blockDim.x + threadIdx.x;
  if (i < NPOS) atomicAdd(&cnt[idx[i]], 1);
}

__global__ void route_offsets_kernel(const int* __restrict__ cnt, int* __restrict__ seg) {
  if (threadIdx.x == 0) {
    int s = 0;
    for (int e = 0; e < NEXP; ++e) {
      seg[e] = s;
      s += (cnt[e] + 15) & ~15;   // pad each expert segment to multiple of 16
    }
    seg[NEXP] = s;                // total padded rows
  }
}

__global__ void route_scatter_kernel(const int* __restrict__ idx, const float* __restrict__ w,
                                     const int* __restrict__ seg, int* __restrict__ cur,
                                     int* __restrict__ tok, float* __restrict__ wgt) {
  int i = blockIdx.x * blockDim.x + threadIdx.x;
  if (i < NPOS) {
    int e = idx[i];
    int p = seg[e] + atomicAdd(&cur[e], 1);
    tok[p] = i / TOPK;
    wgt[p] = w[i];
  }
}

// --------------------------------------------------- gate/up + SiLU pass ---
// Grid: (IDIM/128, CAP/16). Block: 128 threads = 4 waves; each wave owns TWO
// 16x16 output tiles over the intermediate dim (A fragment reused by 4 WMMAs).

__global__ __launch_bounds__(128)
void moe_gateup_kernel(const float* __restrict__ hidden,
                       const float* __restrict__ gate_w,
                       const float* __restrict__ up_w,
                       const int*   __restrict__ seg,
                       const int*   __restrict__ tok,
                       const float* __restrict__ wgt,
                       float*       __restrict__ inter) {
  const int row0 = blockIdx.y * 16;
  if (row0 >= seg[NEXP]) return;          // beyond total padded rows
  int e = 0;
#pragma unroll
  for (int k = 1; k < NEXP; ++k) e += (row0 >= seg[k]) ? 1 : 0;

  const int wave = threadIdx.x >> 5;
  const int lane = threadIdx.x & 31;
  const int lo   = lane & 15;             // A: row m ; B: col n ; D: col n
  const int hi   = lane >> 4;             // selects K pair / D row half
  const int nbase = blockIdx.x * 128 + wave * 32;   // two 16-wide tiles

  // A fragment source: this lane streams token row (m = lo), k-pair 2*hi
  const float* xrow = hidden + (size_t)tok[row0 + lo] * HDIM + 2 * hi;
  // B fragment source: column n = nbase+lo of W[I,H], k-pair 2*hi; second
  // tile at n+16 is a constant 16*HDIM float offset (folds into inst offset).
  const size_t wb = (size_t)e * IDIM * HDIM + (size_t)(nbase + lo) * HDIM + 2 * hi;
  const float* gp = gate_w + wb;
  const float* up = up_w   + wb;

  v8f cg0 = {0.f,0.f,0.f,0.f,0.f,0.f,0.f,0.f};
  v8f cg1 = {0.f,0.f,0.f,0.f,0.f,0.f,0.f,0.f};
  v8f cu0 = {0.f,0.f,0.f,0.f,0.f,0.f,0.f,0.f};
  v8f cu1 = {0.f,0.f,0.f,0.f,0.f,0.f,0.f,0.f};

#pragma unroll 4
  for (int h = 0; h < HDIM; h += 4) {
    v2f a   = *(const v2f*)(xrow + h);
    v2f bg0 = *(const v2f*)(gp + h);
    v2f bg1 = *(const v2f*)(gp + h + 16 * HDIM);
    v2f bu0 = *(const v2f*)(up + h);
    v2f bu1 = *(const v2f*)(up + h + 16 * HDIM);
    cg0 = __builtin_amdgcn_wmma_f32_16x16x4_f32(false, a, false, bg0, (short)0, cg0, false, false);
    cg1 = __builtin_amdgcn_wmma_f32_16x16x4_f32(false, a, false, bg1, (short)0, cg1, false, false);
    cu0 = __builtin_amdgcn_wmma_f32_16x16x4_f32(false, a, false, bu0, (short)0, cu0, false, false);
    cu1 = __builtin_amdgcn_wmma_f32_16x16x4_f32(false, a, false, bu1, (short)0, cu1, false, false);
  }

  // D(m = r + 8*hi, n = lo): fused SiLU * up * routing-weight, store inter.
#pragma unroll
  for (int r = 0; r < 8; ++r) {
    const int m = r + 8 * hi;
    const float cw = wgt[row0 + m];             // 0.0 on pad rows
    float* dst = inter + (size_t)(row0 + m) * IDIM + nbase + lo;
    const float g0 = cg0[r], u0 = cu0[r];
    const float s0 = g0 * __builtin_amdgcn_rcpf(1.0f + __expf(-g0));   // silu
    dst[0] = cw * s0 * u0;
    const float g1 = cg1[r], u1 = cu1[r];
    const float s1 = g1 * __builtin_amdgcn_rcpf(1.0f + __expf(-g1));
    dst[16] = cw * s1 * u1;
  }
}

// --------------------------------------------------------- down-proj pass ---
// Grid: (HDIM/128, CAP/16). D = inter[16xI] * down^T, two 16x16 tiles per
// wave over H, atomic scatter-add into out.

__global__ __launch_bounds__(128)
void moe_down_kernel(const float* __restrict__ inter,
                     const float* __restrict__ down_w,
                     const int*   __restrict__ seg,
                     const int*   __restrict__ tok,
                     float*       __restrict__ out) {
  const int row0 = blockIdx.y * 16;
  if (row0 >= seg[NEXP]) return;
  int e = 0;
#pragma unroll
  for (int k = 1; k < NEXP; ++k) e += (row0 >= seg[k]) ? 1 : 0;

  const int wave = threadIdx.x >> 5;
  const int lane = threadIdx.x & 31;
  const int lo   = lane & 15;
  const int hi   = lane >> 4;
  const int nbase = blockIdx.x * 128 + wave * 32;   // over H, two tiles

  const float* arow = inter + (size_t)(row0 + lo) * IDIM + 2 * hi;
  const float* bp   = down_w + (size_t)e * HDIM * IDIM + (size_t)(nbase + lo) * IDIM + 2 * hi;

  v8f c0 = {0.f,0.f,0.f,0.f,0.f,0.f,0.f,0.f};
  v8f c1 = {0.f,0.f,0.f,0.f,0.f,0.f,0.f,0.f};

#pragma unroll 4
  for (int i = 0; i < IDIM; i += 4) {
    v2f a  = *(const v2f*)(arow + i);
    v2f b0 = *(const v2f*)(bp + i);
    v2f b1 = *(const v2f*)(bp + i + 16 * IDIM);
    c0 = __builtin_amdgcn_wmma_f32_16x16x4_f32(false, a, false, b0, (short)0, c0, false, false);
    c1 = __builtin_amdgcn_wmma_f32_16x16x4_f32(false, a, false, b1, (short)0, c1, false, false);
  }

#pragma unroll
  for (int r = 0; r < 8; ++r) {
    const int m = r + 8 * hi;
    const int t = tok[row0 + m];                  // token 0 on pad rows (adds 0.0)
    float* dst = out + (size_t)t * HDIM + nbase + lo;
    atomicAdd(dst,      c0[r]);
    atomicAdd(dst + 16, c1[r]);
  }
}

// ------------------------------------------------------------------ launch ---

extern "C" void kernel_launch(void* const* d_in, const int* in_sizes, int n_in,
                              void* d_out, int out_size, void* d_ws, size_t ws_size,
                              hipStream_t stream) {
  const float* hidden = (const float*)d_in[0];
  const int*   topk_i = (const int*)d_in[1];
  const float* topk_w = (const float*)d_in[2];
  const float* gate_w = (const float*)d_in[3];
  const float* up_w   = (const float*)d_in[4];
  const float* down_w = (const float*)d_in[5];
  float* out = (float*)d_out;

  char* ws = (char*)d_ws;
  int*   cnt   = (int*)(ws + OFF_COUNT);
  int*   cur   = (int*)(ws + OFF_CURSOR);
  int*   seg   = (int*)(ws + OFF_SEG);
  int*   tok   = (int*)(ws + OFF_TOK);
  float* wgt   = (float*)(ws + OFF_WEIGHT);
  float* inter = (float*)(ws + OFF_INTER);

  zero_out_kernel<<<(T_TOK * HDIM / 4) / 256, 256, 0, stream>>>((float4*)out);
  route_init_kernel<<<(CAP + 255) / 256, 256, 0, stream>>>(cnt, cur, tok, wgt);
  route_count_kernel<<<NPOS / 256, 256, 0, stream>>>(topk_i, cnt);
  route_offsets_kernel<<<1, 32, 0, stream>>>(cnt, seg);
  route_scatter_kernel<<<NPOS / 256, 256, 0, stream>>>(topk_i, topk_w, seg, cur, tok, wgt);

  moe_gateup_kernel<<<dim3(IDIM / 128, CAP / 16), 128, 0, stream>>>(
      hidden, gate_w, up_w, seg, tok, wgt, inter);
  moe_down_kernel<<<dim3(HDIM / 128, CAP / 16), 128, 0, stream>>>(
      inter, down_w, seg, tok, out);
}